// WideBNetModel_1864015806669
// MI455X (gfx1250) — compile-verified
//
#include <hip/hip_runtime.h>

typedef __attribute__((ext_vector_type(2))) float v2f;
typedef __attribute__((ext_vector_type(8))) float v8f;

#if defined(__HIP_DEVICE_COMPILE__)
#if !__has_builtin(__builtin_amdgcn_wmma_f32_16x16x4_f32)
#error "gfx1250 f32 WMMA builtin __builtin_amdgcn_wmma_f32_16x16x4_f32 not available"
#endif
#endif

// D = A(16x4,f32) * B(4x16,f32) + C(16x16,f32)  -- CDNA5 fp32 matrix path
__device__ __forceinline__ v8f wmma4(v2f a, v2f b, v8f c) {
#if defined(__HIP_DEVICE_COMPILE__)
  return __builtin_amdgcn_wmma_f32_16x16x4_f32(
      /*neg_a=*/false, a, /*neg_b=*/false, b,
      /*c_mod=*/(short)0, c, /*reuse_a=*/false, /*reuse_b=*/false);
#else
  (void)a; (void)b;
  return c;  // host pass never executes device code; stub for parsing only
#endif
}

// ---------------------------------------------------------------------------
// Complex block-GEMM:  yr = xr@r1 + xi@i1 ; yi = xr@i2 + xi@r2, per block n.
// One wave (32 lanes) computes one 16(batch) x 16(out-col) tile for one block.
// All reshapes / permutations / concats / repeats of the reference are folded
// into the epilogue address computation (all perms for L=4 are involutions:
// perm(4,3)=identity, perm(4,2)=4x4 transpose in 16-blocks, switch=16x16 T).
// ---------------------------------------------------------------------------
constexpr int M_V0 = 0, M_V1 = 1, M_V2 = 2, M_H3 = 3, M_H2 = 4, M_MA = 5,
              M_MB_ADD = 6, M_MB_LAST = 7, M_G2 = 8, M_G3 = 9, M_U = 10;

__device__ __forceinline__ int perm2_map(int n) {  // blockwise 4x4 transpose
  int t = n & 15;
  return (n & ~15) | ((t & 3) << 2) | (t >> 2);
}
__device__ __forceinline__ int sw_map(int n) {     // 16x16 transpose of 256
  return ((n & 15) << 4) | (n >> 4);
}

template <int MODE>
__global__ __launch_bounds__(32) void cgemm(
    const float* __restrict__ A, const float* __restrict__ r1,
    const float* __restrict__ i1, const float* __restrict__ r2,
    const float* __restrict__ i2, float* __restrict__ out) {
  constexpr int J = (MODE == M_V0)   ? 64
                    : (MODE == M_V1) ? 256
                    : (MODE == M_V2) ? 1024
                    : (MODE == M_H3) ? 128
                    : (MODE == M_H2 || MODE == M_G2 || MODE == M_G3) ? 192
                                                                     : 48;
  constexpr int K = (MODE <= M_V2)   ? 16
                    : (MODE == M_H3) ? 128
                    : (MODE == M_H2 || MODE == M_G2 || MODE == M_G3) ? 192
                    : (MODE == M_U) ? 64
                                    : 48;
  constexpr bool AIN  = (MODE <= M_V2);   // A from raw inputs (stride-3 scales)
  constexpr bool CPLX = (MODE != M_U);    // U only needs the real part
  constexpr int ATOT  = (MODE == M_H3) ? 8192 : 12288;  // slab per (b,part)
  constexpr int NT = K / 16;

  const int lane = threadIdx.x;
  const int lcol = lane & 15;
  const int half = lane >> 4;
  const int koff = half << 1;

  const int tile = blockIdx.x;
  const int n  = tile / (NT * 4);
  const int mt = (tile / NT) & 3;
  const int nt = tile % NT;

  const int mrow = (mt << 4) + lcol;   // batch row this lane supplies to A
  const int ncol = nt * 16 + lcol;     // output column this lane supplies to B

  const size_t wofs = (size_t)n * J * K;
  const float* wr1 = r1 + wofs;
  const float* wi1 = i1 + wofs;
  const float* wr2 = r2 + wofs;
  const float* wi2 = i2 + wofs;

  size_t aR, aI;
  if constexpr (AIN) {
    aR = ((size_t)(mrow * 2 + 0) * 16384 + (size_t)n * J) * 3;
    aI = ((size_t)(mrow * 2 + 1) * 16384 + (size_t)n * J) * 3;
  } else {
    aR = (size_t)(mrow * 2 + 0) * ATOT + (size_t)n * J;
    aI = (size_t)(mrow * 2 + 1) * ATOT + (size_t)n * J;
  }

  v8f accR = {0.f, 0.f, 0.f, 0.f, 0.f, 0.f, 0.f, 0.f};
  v8f accI = {0.f, 0.f, 0.f, 0.f, 0.f, 0.f, 0.f, 0.f};

  for (int k0 = 0; k0 < J; k0 += 4) {
    const int j = k0 + koff;  // ISA A layout: lane-half -> K+{0,2}, vgpr -> K+1
    v2f ar, ai;
    if constexpr (AIN) {
      ar.x = A[aR + (size_t)j * 3];
      ar.y = A[aR + (size_t)(j + 1) * 3];
      ai.x = A[aI + (size_t)j * 3];
      ai.y = A[aI + (size_t)(j + 1) * 3];
    } else {
      ar.x = A[aR + j];
      ar.y = A[aR + j + 1];
      ai.x = A[aI + j];
      ai.y = A[aI + j + 1];
    }
    const size_t wb = (size_t)j * K + ncol;
    v2f b;
    b.x = wr1[wb]; b.y = wr1[wb + K]; accR = wmma4(ar, b, accR);
    b.x = wi1[wb]; b.y = wi1[wb + K]; accR = wmma4(ai, b, accR);
    if constexpr (CPLX) {
      b.x = wi2[wb]; b.y = wi2[wb + K]; accI = wmma4(ar, b, accI);
      b.x = wr2[wb]; b.y = wr2[wb + K]; accI = wmma4(ai, b, accI);
    }
  }

  // C layout: vgpr r holds M=r (lanes 0-15) / M=r+8 (lanes 16-31), N=lane&15.
#pragma unroll
  for (int r = 0; r < 8; ++r) {
    const int brow = (mt << 4) + r + (half << 3);
    const int kcol = ncol;
    const float vr = accR[r];
    const float vi = accI[r];
    const size_t b0 = (size_t)(brow * 2 + 0) * 256;
    const size_t b1 = (size_t)(brow * 2 + 1) * 256;
    if constexpr (MODE == M_V0) {
      out[(b0 + n) * 32 + kcol] = vr;
      out[(b1 + n) * 32 + kcol] = vi;
    } else if constexpr (MODE == M_V1) {   // repeat 4x along n, concat @16
#pragma unroll
      for (int t = 0; t < 4; ++t) {
        out[(b0 + n * 4 + t) * 32 + 16 + kcol] = vr;
        out[(b1 + n * 4 + t) * 32 + 16 + kcol] = vi;
      }
    } else if constexpr (MODE == M_V2) {   // repeat 16x along n, concat @32
#pragma unroll
      for (int t = 0; t < 16; ++t) {
        out[(b0 + n * 16 + t) * 48 + 32 + kcol] = vr;
        out[(b1 + n * 16 + t) * 48 + 32 + kcol] = vi;
      }
    } else if constexpr (MODE == M_H3) {   // reshape (64,128)->(256,32)+perm2
      const int ng = 4 * n + (kcol >> 5);
      const int c = kcol & 31;
      const int q = perm2_map(ng);
      out[(b0 + q) * 48 + c] = vr;
      out[(b1 + q) * 48 + c] = vi;
    } else if constexpr (MODE == M_H2 || MODE == M_G2 || MODE == M_G3) {
      const int ng = 4 * n + kcol / 48;    // reshape (64,192)->(256,48)
      const int c = kcol % 48;
      const int q = (MODE == M_H2)   ? sw_map(ng)
                    : (MODE == M_G2) ? perm2_map(ng)
                                     : ng;
      out[(b0 + q) * 48 + c] = vr;
      out[(b1 + q) * 48 + c] = vi;
    } else if constexpr (MODE == M_MA) {   // relu epilogue
      out[(b0 + n) * 48 + kcol] = fmaxf(vr, 0.f);
      out[(b1 + n) * 48 + kcol] = fmaxf(vi, 0.f);
    } else if constexpr (MODE == M_MB_ADD) {  // in-place residual add
      out[(b0 + n) * 48 + kcol] += vr;
      out[(b1 + n) * 48 + kcol] += vi;
    } else if constexpr (MODE == M_MB_LAST) {
      out[(b0 + n) * 48 + kcol] = vr;
      out[(b1 + n) * 48 + kcol] = vi;
    } else if constexpr (MODE == M_U) {    // real part only, (B,16384)
      out[(size_t)brow * 16384 + n * 64 + kcol] = vr;
    }
  }
}

// ---------------------------------------------------------------------------
// Morton gather + fused DenseNet conv stack (5x conv3x3+relu+concat, final
// conv3x3).  8x8 output tile, 20x20 halo region, all 31 channels resident in
// LDS (31*400*4 = 49.6 KB).  Out-of-image positions are forced to zero so the
// SAME zero-padding semantics of the reference survive tiling.
// ---------------------------------------------------------------------------
constexpr int CT = 8;            // output tile side
constexpr int CR = CT + 12;      // 20 : input region side (6-pixel halo)
constexpr int CA = CR * CR;      // 400

__global__ __launch_bounds__(256) void fused_conv(
    const float* __restrict__ Ur, const int* __restrict__ mort,
    const float* __restrict__ w0, const float* __restrict__ bb0,
    const float* __restrict__ w1, const float* __restrict__ bb1,
    const float* __restrict__ w2, const float* __restrict__ bb2,
    const float* __restrict__ w3, const float* __restrict__ bb3,
    const float* __restrict__ w4, const float* __restrict__ bb4,
    const float* __restrict__ wf, const float* __restrict__ bf,
    float* __restrict__ out) {
  __shared__ float lds[31 * CA];
  const int tid = threadIdx.x;
  const int tx = blockIdx.x, ty = blockIdx.y, bimg = blockIdx.z;
  const int gx0 = tx * CT - 6, gy0 = ty * CT - 6;

  // channel 0: Morton-gathered U output, zero outside image
  for (int p = tid; p < CA; p += 256) {
    const int ly = p / CR, lx = p % CR;
    const int gy = gy0 + ly, gx = gx0 + lx;
    float v = 0.f;
    if (gy >= 0 && gy < 128 && gx >= 0 && gx < 128)
      v = Ur[(size_t)bimg * 16384 + mort[gy * 128 + gx]];
    lds[p] = v;
  }
  __syncthreads();

  const float* ws[5] = {w0, w1, w2, w3, w4};
  const float* bs[5] = {bb0, bb1, bb2, bb3, bb4};
  int cin = 1;
  for (int s = 0; s < 5; ++s) {
    const float* w = ws[s];
    const float* bia = bs[s];
    const int lo = s + 1;
    const int R = CR - 2 - 2 * s;  // shrinking valid region per stage
    for (int p = tid; p < R * R; p += 256) {
      const int oy = lo + p / R, ox = lo + p % R;
      float a0 = bia[0], a1 = bia[1], a2 = bia[2];
      float a3 = bia[3], a4 = bia[4], a5 = bia[5];
      for (int c = 0; c < cin; ++c) {
        const float* pl = &lds[c * CA + (oy - 1) * CR + (ox - 1)];
#pragma unroll
        for (int dy = 0; dy < 3; ++dy) {
#pragma unroll
          for (int dx = 0; dx < 3; ++dx) {
            const float x = pl[dy * CR + dx];
            const float* wp = &w[((dy * 3 + dx) * cin + c) * 6];  // HWIO
            a0 += x * wp[0]; a1 += x * wp[1]; a2 += x * wp[2];
            a3 += x * wp[3]; a4 += x * wp[4]; a5 += x * wp[5];
          }
        }
      }
      const int gy = gy0 + oy, gx = gx0 + ox;
      const bool in = (gy >= 0 && gy < 128 && gx >= 0 && gx < 128);
      const int o = oy * CR + ox;
      lds[(cin + 0) * CA + o] = in ? fmaxf(a0, 0.f) : 0.f;
      lds[(cin + 1) * CA + o] = in ? fmaxf(a1, 0.f) : 0.f;
      lds[(cin + 2) * CA + o] = in ? fmaxf(a2, 0.f) : 0.f;
      lds[(cin + 3) * CA + o] = in ? fmaxf(a3, 0.f) : 0.f;
      lds[(cin + 4) * CA + o] = in ? fmaxf(a4, 0.f) : 0.f;
      lds[(cin + 5) * CA + o] = in ? fmaxf(a5, 0.f) : 0.f;
    }
    __syncthreads();
    cin += 6;
  }

  if (tid < CT * CT) {  // final 31->1 conv, no relu
    const int oy = 6 + tid / CT, ox = 6 + tid % CT;
    float acc = bf[0];
    for (int c = 0; c < 31; ++c) {
      const float* pl = &lds[c * CA + (oy - 1) * CR + (ox - 1)];
#pragma unroll
      for (int dy = 0; dy < 3; ++dy)
#pragma unroll
        for (int dx = 0; dx < 3; ++dx)
          acc += pl[dy * CR + dx] * wf[(dy * 3 + dx) * 31 + c];
    }
    const int gy = gy0 + oy, gx = gx0 + ox;
    out[(size_t)bimg * 16384 + gy * 128 + gx] = acc;
  }
}

// ---------------------------------------------------------------------------
// Host orchestration.  Auto-detects the flattening order of the nested params
// dict (insertion order vs jax-pytree sorted keys) from in_sizes.
// ---------------------------------------------------------------------------
extern "C" void kernel_launch(void* const* d_in, const int* in_sizes, int n_in,
                              void* d_out, int out_size, void* d_ws,
                              size_t ws_size, hipStream_t stream) {
  auto F = [&](int i) -> const float* { return (const float*)d_in[i]; };

  const bool topSorted = (in_sizes[0] == 16384);  // idx_morton first
  const int inputsPos = topSorted ? 1 : 0;
  const int P = topSorted ? 2 : 1;                // first params leaf
  const bool sortedLeaves = topSorted || (in_sizes[P] == 64 * 192 * 192);
  const int idxPos = topSorted ? 0 : (n_in - 1);

  struct CB { const float *r1, *i1, *r2, *i2; };
  auto cbAt = [&](int s) -> CB {
    const float *p0 = F(s), *p1 = F(s + 1), *p2 = F(s + 2), *p3 = F(s + 3);
    if (!sortedLeaves) return CB{p0, p1, p2, p3};  // r1,i1,r2,i2
    return CB{p2, p0, p3, p1};                     // stored as i1,i2,r1,r2
  };

  CB V0, V1, V2, H3, H2, MA[3], MB[3], G2, G3, UU;
  const float *cw[5], *cbi[5], *fw, *fb;
  if (!sortedLeaves) {
    V0 = cbAt(P + 0); V1 = cbAt(P + 4); V2 = cbAt(P + 8);
    H3 = cbAt(P + 12); H2 = cbAt(P + 16);
    for (int m = 0; m < 3; ++m) { MA[m] = cbAt(P + 20 + 8 * m); MB[m] = cbAt(P + 24 + 8 * m); }
    G2 = cbAt(P + 44); G3 = cbAt(P + 48); UU = cbAt(P + 52);
    const int q = P + 56;
    for (int i = 0; i < 5; ++i) { cw[i] = F(q + 2 * i); cbi[i] = F(q + 2 * i + 1); }
    fw = F(q + 10); fb = F(q + 11);
  } else {  // ASCII-sorted keys: G2,G3,H2,H3,M*,U,V*, then conv*_b/w, final_b/w
    G2 = cbAt(P + 0); G3 = cbAt(P + 4); H2 = cbAt(P + 8); H3 = cbAt(P + 12);
    for (int m = 0; m < 3; ++m) { MA[m] = cbAt(P + 16 + 8 * m); MB[m] = cbAt(P + 20 + 8 * m); }
    UU = cbAt(P + 40); V0 = cbAt(P + 44); V1 = cbAt(P + 48); V2 = cbAt(P + 52);
    const int q = P + 56;
    for (int i = 0; i < 5; ++i) { cbi[i] = F(q + 2 * i); cw[i] = F(q + 2 * i + 1); }
    fb = F(q + 10); fw = F(q + 11);
  }
  const float* inp = F(inputsPos);
  const int* mort = (const int*)d_in[idxPos];

  // workspace layout (floats):  B1 | Y2 | Y | T | Ub   (~27.3 MB total)
  float* wsf = (float*)d_ws;
  float* B1 = wsf;                 // (64,2,256,32)
  float* Y2 = B1 + 1048576;        // (64,2,256,48)
  float* Y  = Y2 + 1572864;        // (64,2,256,48)
  float* T  = Y  + 1572864;        // (64,2,256,48)
  float* Ub = T  + 1572864;        // (64,16384)

  // grid = NB * (K/16) * 4 m-tiles, one wave per workgroup
  cgemm<M_V0><<<1024, 32, 0, stream>>>(inp + 0, V0.r1, V0.i1, V0.r2, V0.i2, B1);
  cgemm<M_V1><<<256, 32, 0, stream>>>(inp + 1, V1.r1, V1.i1, V1.r2, V1.i2, B1);
  cgemm<M_V2><<<64, 32, 0, stream>>>(inp + 2, V2.r1, V2.i1, V2.r2, V2.i2, Y2);
  cgemm<M_H3><<<2048, 32, 0, stream>>>(B1, H3.r1, H3.i1, H3.r2, H3.i2, Y2);
  cgemm<M_H2><<<3072, 32, 0, stream>>>(Y2, H2.r1, H2.i1, H2.r2, H2.i2, Y);
  for (int m = 0; m < 3; ++m) {
    cgemm<M_MA><<<3072, 32, 0, stream>>>(Y, MA[m].r1, MA[m].i1, MA[m].r2, MA[m].i2, T);
    if (m < 2)
      cgemm<M_MB_ADD><<<3072, 32, 0, stream>>>(T, MB[m].r1, MB[m].i1, MB[m].r2, MB[m].i2, Y);
    else
      cgemm<M_MB_LAST><<<3072, 32, 0, stream>>>(T, MB[m].r1, MB[m].i1, MB[m].r2, MB[m].i2, Y);
  }
  cgemm<M_G2><<<3072, 32, 0, stream>>>(Y, G2.r1, G2.i1, G2.r2, G2.i2, Y2);
  cgemm<M_G3><<<3072, 32, 0, stream>>>(Y2, G3.r1, G3.i1, G3.r2, G3.i2, Y);
  cgemm<M_U><<<4096, 32, 0, stream>>>(Y, UU.r1, UU.i1, UU.r2, UU.i2, Ub);

  fused_conv<<<dim3(16, 16, 64), 256, 0, stream>>>(
      Ub, mort, cw[0], cbi[0], cw[1], cbi[1], cw[2], cbi[2], cw[3], cbi[3],
      cw[4], cbi[4], fw, fb, (float*)d_out);

  (void)out_size; (void)ws_size;
}